// JEHierarchicalClassifier_66769561584339
// MI455X (gfx1250) — compile-verified
//
#include <hip/hip_runtime.h>

// ---------------------------------------------------------------------------
// JEHierarchicalClassifier on MI455X (gfx1250, wave32, WMMA f32 16x16x4)
//
// Shapes: B=8192, Din=2048, D=1024, P=16, C0=32, C1=64
// d_out layout (floats): parent_logits[8192*16] | child0[8192*32] |
//                        child1[8192*64] | parent_emb[8192*1024] |
//                        child_emb[8192*1024]
// d_ws: parent_class int[8192]
// ---------------------------------------------------------------------------

typedef float v2f __attribute__((ext_vector_type(2)));
typedef float v8f __attribute__((ext_vector_type(8)));

#define LN_EPS 1e-5f

__device__ __forceinline__ float wave_sum32(float v) {
    v += __shfl_xor(v, 16, 32);
    v += __shfl_xor(v, 8, 32);
    v += __shfl_xor(v, 4, 32);
    v += __shfl_xor(v, 2, 32);
    v += __shfl_xor(v, 1, 32);
    return v;
}

// reduction within each 16-lane half of a wave32 (xor masks < 16 never cross)
__device__ __forceinline__ float half_sum16(float v) {
    v += __shfl_xor(v, 8, 32);
    v += __shfl_xor(v, 4, 32);
    v += __shfl_xor(v, 2, 32);
    v += __shfl_xor(v, 1, 32);
    return v;
}

// ---------------------------------------------------------------------------
// Kernel 1: x[8192,2048] @ W[2048,1024]  (z=0: parent_proj_w, z=1: child_proj_w)
// Block: 256 threads = 8 waves. Tile: 256(M) x 64(N), K chunked by 32 via LDS.
// Each wave: 32x64 strip = 2 A frags x 4 B frags -> 8 WMMA per k-step
// (6 LDS b64 loads per 8 WMMAs).
//
// LDS layouts:
//  xs : row-major, row stride 36 floats (pad kills the stride-32 bank alias,
//       keeps b64/b128 alignment).
//  wsm: paired-K: element (k,n) at wsm[(k>>1)*160 + n*2 + (k&1)], so a B
//       fragment {W[k][n], W[k+1][n]} is ONE aligned ds_load_b64 -> no
//       v_dual_mov repacking in front of the WMMAs. kp-stride 160 puts the
//       two half-waves on disjoint even-bank sets.
// ---------------------------------------------------------------------------
#define XS_STRIDE 36
#define WS_STRIDE 160

__global__ __launch_bounds__(256) void jh_gemm_dual(
    const float* __restrict__ x,
    const float* __restrict__ w_parent,
    const float* __restrict__ w_child,
    float* __restrict__ out_parent,
    float* __restrict__ out_child)
{
    const int K = 2048, N = 1024;
    const float* __restrict__ W   = (blockIdx.z == 0) ? w_parent  : w_child;
    float* __restrict__       out = (blockIdx.z == 0) ? out_parent : out_child;
    const int n0 = blockIdx.x * 64;
    const int m0 = blockIdx.y * 256;

    __shared__ float xs[256 * XS_STRIDE];   // 36 KB
    __shared__ float wsm[16 * WS_STRIDE];   // 10 KB

    const int tid  = threadIdx.x;
    const int wave = tid >> 5;
    const int lane = tid & 31;
    const int half = lane >> 4;
    const int m    = lane & 15;

    v8f acc[2][4];
#pragma unroll
    for (int f = 0; f < 2; ++f)
#pragma unroll
        for (int t = 0; t < 4; ++t)
#pragma unroll
            for (int i = 0; i < 8; ++i) acc[f][t][i] = 0.0f;

    for (int k0 = 0; k0 < K; k0 += 32) {
        // cooperative load: x tile 256x32 (2048 float4, 8/thread)
#pragma unroll
        for (int i = 0; i < 8; ++i) {
            int idx = tid + 256 * i;
            int r = idx >> 3, c = (idx & 7) << 2;
            float4 v = *(const float4*)(x + (size_t)(m0 + r) * K + k0 + c);
            *(float4*)(xs + r * XS_STRIDE + c) = v;
        }
        // cooperative load: w tile 32x64 into paired-K layout (2/thread)
#pragma unroll
        for (int i = 0; i < 2; ++i) {
            int idx = tid + 256 * i;
            int kr = idx >> 4, c = (idx & 15) << 2;
            float4 v = *(const float4*)(W + (size_t)(k0 + kr) * N + n0 + c);
            float* dst = wsm + (kr >> 1) * WS_STRIDE + (kr & 1);
            dst[(c + 0) * 2] = v.x;
            dst[(c + 1) * 2] = v.y;
            dst[(c + 2) * 2] = v.z;
            dst[(c + 3) * 2] = v.w;
        }
        // prefetch next K-chunk into cache hierarchy (global_prefetch_b8)
        if (k0 + 32 < K) {
            __builtin_prefetch(x + (size_t)(m0 + tid) * K + (k0 + 32), 0, 3);
            __builtin_prefetch(W + (size_t)(k0 + 32 + (tid >> 3)) * N + n0 + (tid & 7) * 8, 0, 3);
        }
        __syncthreads();

        const float* xr0 = xs + (wave * 32 + m) * XS_STRIDE;
        const float* xr1 = xr0 + 16 * XS_STRIDE;
#pragma unroll
        for (int kk = 0; kk < 32; kk += 4) {
            // A frag 16x4: lanes0-15 -> K pair (kk,kk+1); lanes16-31 -> (kk+2,kk+3)
            const int kb = kk + half * 2;
            v2f a0 = *(const v2f*)(xr0 + kb);
            v2f a1 = *(const v2f*)(xr1 + kb);
            const float* wp = wsm + (kb >> 1) * WS_STRIDE + m * 2;
#pragma unroll
            for (int t = 0; t < 4; ++t) {
                v2f b = *(const v2f*)(wp + t * 32);   // {W[kb][n], W[kb+1][n]}
                acc[0][t] = __builtin_amdgcn_wmma_f32_16x16x4_f32(
                    false, a0, false, b, (short)0, acc[0][t], false, false);
                acc[1][t] = __builtin_amdgcn_wmma_f32_16x16x4_f32(
                    false, a1, false, b, (short)0, acc[1][t], false, false);
            }
        }
        __syncthreads();
    }

    // C layout: VGPR r -> row r (lanes 0-15) / row r+8 (lanes 16-31), col = lane&15
#pragma unroll
    for (int f = 0; f < 2; ++f)
#pragma unroll
        for (int t = 0; t < 4; ++t)
#pragma unroll
            for (int r = 0; r < 8; ++r) {
                int row = m0 + wave * 32 + f * 16 + r + half * 8;
                int col = n0 + t * 16 + m;
                out[(size_t)row * N + col] = acc[f][t][r];
            }
}

// ---------------------------------------------------------------------------
// Kernel 2: y = LN(raw + bias) in place over D=1024. One wave32 per row.
// grid: (1024, 2), block 256 (8 waves)
// ---------------------------------------------------------------------------
__global__ __launch_bounds__(256) void jh_ln_bias(
    float* __restrict__ emb_parent, float* __restrict__ emb_child,
    const float* __restrict__ b_parent, const float* __restrict__ b_child)
{
    const int D = 1024;
    float* base        = (blockIdx.y == 0) ? emb_parent : emb_child;
    const float* bias  = (blockIdx.y == 0) ? b_parent   : b_child;
    const int wave = threadIdx.x >> 5;
    const int lane = threadIdx.x & 31;
    const int row  = blockIdx.x * 8 + wave;
    float* p = base + (size_t)row * D;

    float4 v[8];
    float s = 0.f, s2 = 0.f;
#pragma unroll
    for (int j = 0; j < 8; ++j) {
        int c4 = lane + 32 * j;
        float4 t  = ((const float4*)p)[c4];
        float4 bb = ((const float4*)bias)[c4];
        t.x += bb.x; t.y += bb.y; t.z += bb.z; t.w += bb.w;
        v[j] = t;
        s  += t.x + t.y + t.z + t.w;
        s2 += t.x * t.x + t.y * t.y + t.z * t.z + t.w * t.w;
    }
    s  = wave_sum32(s);
    s2 = wave_sum32(s2);
    const float mean = s * (1.0f / 1024.0f);
    const float var  = s2 * (1.0f / 1024.0f) - mean * mean;
    const float rstd = rsqrtf(var + LN_EPS);
#pragma unroll
    for (int j = 0; j < 8; ++j) {
        float4 t = v[j];
        t.x = (t.x - mean) * rstd;
        t.y = (t.y - mean) * rstd;
        t.z = (t.z - mean) * rstd;
        t.w = (t.w - mean) * rstd;
        ((float4*)p)[lane + 32 * j] = t;
    }
}

// ---------------------------------------------------------------------------
// Kernel 3: parent_logits = LN(pemb @ Wcls + b), argmax -> parent_class.
// One wave per 16 rows; WMMA f32 16x16x4 over K=1024. N=16 LN fits a half-wave.
// ---------------------------------------------------------------------------
__global__ __launch_bounds__(32) void jh_parent_cls(
    const float* __restrict__ pemb,  // [8192,1024] (normalized)
    const float* __restrict__ wcls,  // [1024,16]
    const float* __restrict__ bcls,  // [16]
    float* __restrict__ plog,        // [8192,16]
    int* __restrict__ pclass)        // [8192]
{
    const int D = 1024, P = 16;
    const int lane = threadIdx.x & 31;
    const int half = lane >> 4;
    const int m    = lane & 15;
    const int row0 = blockIdx.x * 16;

    v8f acc;
#pragma unroll
    for (int i = 0; i < 8; ++i) acc[i] = 0.0f;

    const float* arow = pemb + (size_t)(row0 + m) * D;
    for (int k0 = 0; k0 < D; k0 += 4) {
        const int ak = k0 + half * 2;
        v2f a; a.x = arow[ak];          a.y = arow[ak + 1];
        v2f b; b.x = wcls[ak * P + m];  b.y = wcls[(ak + 1) * P + m];
        acc = __builtin_amdgcn_wmma_f32_16x16x4_f32(
            false, a, false, b, (short)0, acc, false, false);
    }

    const float bv = bcls[m];
#pragma unroll
    for (int r = 0; r < 8; ++r) {
        const int row = row0 + r + half * 8;
        float val = acc[r] + bv;
        // LN over the 16 logits of this row (spread over one half-wave)
        const float mean = half_sum16(val) * (1.0f / 16.0f);
        const float d    = val - mean;
        const float var  = half_sum16(d * d) * (1.0f / 16.0f);
        const float y    = d * rsqrtf(var + LN_EPS);
        plog[(size_t)row * P + m] = y;
        // argmax over N=16 (LN is a positive-scale affine map -> same argmax)
        float best = y; int bidx = m;
#pragma unroll
        for (int msk = 8; msk >= 1; msk >>= 1) {
            float ob = __shfl_xor(best, msk, 32);
            int   oi = __shfl_xor(bidx, msk, 32);
            if (ob > best || (ob == best && oi < bidx)) { best = ob; bidx = oi; }
        }
        if (m == 0) pclass[row] = bidx;
    }
}

// ---------------------------------------------------------------------------
// Kernel 4: gathered child heads. One wave per sample: only the SELECTED
// parent's head is computed (vs. reference's dense-all-parents + select).
// lane -> column; C0=32 (1 acc), C1=64 (2 accs); LN via wave reductions.
// ---------------------------------------------------------------------------
__global__ __launch_bounds__(256) void jh_child_heads(
    const float* __restrict__ cemb,  // [8192,1024] (normalized)
    const float* __restrict__ w0,    // [16,1024,32]
    const float* __restrict__ b0,    // [16,32]
    const float* __restrict__ w1,    // [16,1024,64]
    const float* __restrict__ b1,    // [16,64]
    const int* __restrict__ pclass,  // [8192]
    float* __restrict__ out0,        // [8192,32]
    float* __restrict__ out1)        // [8192,64]
{
    const int D = 1024;
    const int wave = threadIdx.x >> 5;
    const int lane = threadIdx.x & 31;
    const int s = blockIdx.x * 8 + wave;
    const int p = pclass[s];

    const float* ce = cemb + (size_t)s * D;
    const float* W0 = w0 + (size_t)p * D * 32;
    const float* W1 = w1 + (size_t)p * D * 64;

    float a0 = 0.f, a1 = 0.f, a2 = 0.f;
    for (int k = 0; k < D; k += 4) {
        const float4 c4 = *(const float4*)(ce + k);
        a0 += c4.x * W0[(k + 0) * 32 + lane] + c4.y * W0[(k + 1) * 32 + lane]
            + c4.z * W0[(k + 2) * 32 + lane] + c4.w * W0[(k + 3) * 32 + lane];
        a1 += c4.x * W1[(k + 0) * 64 + lane] + c4.y * W1[(k + 1) * 64 + lane]
            + c4.z * W1[(k + 2) * 64 + lane] + c4.w * W1[(k + 3) * 64 + lane];
        a2 += c4.x * W1[(k + 0) * 64 + 32 + lane] + c4.y * W1[(k + 1) * 64 + 32 + lane]
            + c4.z * W1[(k + 2) * 64 + 32 + lane] + c4.w * W1[(k + 3) * 64 + 32 + lane];
    }

    // child0: LN over 32 (full wave)
    {
        const float v    = a0 + b0[p * 32 + lane];
        const float mean = wave_sum32(v) * (1.0f / 32.0f);
        const float d    = v - mean;
        const float var  = wave_sum32(d * d) * (1.0f / 32.0f);
        out0[(size_t)s * 32 + lane] = d * rsqrtf(var + LN_EPS);
    }
    // child1: LN over 64 (2 values per lane)
    {
        const float u0   = a1 + b1[p * 64 + lane];
        const float u1   = a2 + b1[p * 64 + 32 + lane];
        const float mean = wave_sum32(u0 + u1) * (1.0f / 64.0f);
        const float d0 = u0 - mean, d1 = u1 - mean;
        const float var  = wave_sum32(d0 * d0 + d1 * d1) * (1.0f / 64.0f);
        const float rstd = rsqrtf(var + LN_EPS);
        out1[(size_t)s * 64 + lane]      = d0 * rstd;
        out1[(size_t)s * 64 + 32 + lane] = d1 * rstd;
    }
}

// ---------------------------------------------------------------------------
extern "C" void kernel_launch(void* const* d_in, const int* in_sizes, int n_in,
                              void* d_out, int out_size, void* d_ws, size_t ws_size,
                              hipStream_t stream) {
    (void)in_sizes; (void)n_in; (void)out_size; (void)ws_size;
    const float* x   = (const float*)d_in[0];
    const float* ppw = (const float*)d_in[1];
    const float* ppb = (const float*)d_in[2];
    const float* cpw = (const float*)d_in[3];
    const float* cpb = (const float*)d_in[4];
    const float* pcw = (const float*)d_in[5];
    const float* pcb = (const float*)d_in[6];
    const float* cw0 = (const float*)d_in[7];
    const float* cb0 = (const float*)d_in[8];
    const float* cw1 = (const float*)d_in[9];
    const float* cb1 = (const float*)d_in[10];

    float* out   = (float*)d_out;
    float* plog  = out;                       // [8192,16]
    float* c0    = plog + 8192 * 16;          // [8192,32]
    float* c1    = c0   + 8192 * 32;          // [8192,64]
    float* pemb  = c1   + 8192 * 64;          // [8192,1024]
    float* cemb  = pemb + 8192 * 1024;        // [8192,1024]
    int*   pclass = (int*)d_ws;               // [8192]

    // 1) both big GEMMs (raw, pre-bias) into d_out emb regions
    dim3 g1(1024 / 64, 8192 / 256, 2);
    jh_gemm_dual<<<g1, 256, 0, stream>>>(x, ppw, cpw, pemb, cemb);

    // 2) bias + LayerNorm in place
    dim3 g2(8192 / 8, 2);
    jh_ln_bias<<<g2, 256, 0, stream>>>(pemb, cemb, ppb, cpb);

    // 3) parent logits (WMMA) + LN + argmax
    jh_parent_cls<<<8192 / 16, 32, 0, stream>>>(pemb, pcw, pcb, plog, pclass);

    // 4) gathered child heads + LN
    jh_child_heads<<<8192 / 8, 256, 0, stream>>>(cemb, cw0, cb0, cw1, cb1,
                                                 pclass, c0, c1);
}